// AttentionBlock_20375324852770
// MI455X (gfx1250) — compile-verified
//
#include <hip/hip_runtime.h>
#include <math.h>

typedef __attribute__((ext_vector_type(2))) float v2f;
typedef __attribute__((ext_vector_type(8))) float v8f;
typedef int v4i __attribute__((vector_size(16)));

#define B_   16
#define L_   1024
#define C_   512
#define G_   32
#define NH_  8
#define D_   64
#define C3_  1536
#define FACTOR 0.35355339059327373f   /* 1/sqrt(sqrt(64)) */

#if __has_builtin(__builtin_amdgcn_global_load_async_to_lds_b128)
#define HAS_ASYNC_LDS 1
typedef __attribute__((address_space(1))) v4i* g_v4i_p;
typedef __attribute__((address_space(3))) v4i* l_v4i_p;
#else
#define HAS_ASYNC_LDS 0
#endif

// Copy 16 bytes global -> LDS.  Async-DMA (ASYNCcnt) when available: no VGPR
// staging, no loadcnt stall in the issuing wave.
__device__ __forceinline__ void copy16_g2l(const float* __restrict__ g,
                                           float* l) {
#if HAS_ASYNC_LDS
  __builtin_amdgcn_global_load_async_to_lds_b128((g_v4i_p)g, (l_v4i_p)l, 0, 0);
#else
  *(float4*)l = *(const float4*)g;
#endif
}

__device__ __forceinline__ void async_wait_all() {
#if HAS_ASYNC_LDS
#if __has_builtin(__builtin_amdgcn_s_wait_asynccnt)
  __builtin_amdgcn_s_wait_asynccnt(0);
#else
  asm volatile("s_wait_asynccnt 0" ::: "memory");
#endif
#endif
}

__device__ __forceinline__ v8f wmma4(v2f a, v2f b, v8f c) {
  // D = A(16x4,f32) * B(4x16,f32) + C(16x16,f32) on the WMMA pipe
  return __builtin_amdgcn_wmma_f32_16x16x4_f32(false, a, false, b, (short)0, c,
                                               false, false);
}

// ---------------------------------------------------------------------------
// GroupNorm statistics: one block per (batch, group); reduce over L*16 = 16384
// ---------------------------------------------------------------------------
__global__ void __launch_bounds__(256) gn_stats(const float* __restrict__ x,
                                                float* __restrict__ meanv,
                                                float* __restrict__ rstdv) {
  const int bg = blockIdx.x;
  const int b = bg >> 5, g = bg & 31;
  const float* xp = x + (size_t)b * L_ * C_ + g * 16;
  const int t = threadIdx.x;
  const int c = t & 15, r0 = t >> 4;
  float s = 0.f, s2 = 0.f;
#pragma unroll 4
  for (int i = 0; i < 64; ++i) {
    float v = xp[(size_t)(r0 + 16 * i) * C_ + c];
    s += v;
    s2 += v * v;
  }
  __shared__ float sh[256], sh2[256];
  sh[t] = s;
  sh2[t] = s2;
  __syncthreads();
  for (int off = 128; off > 0; off >>= 1) {
    if (t < off) { sh[t] += sh[t + off]; sh2[t] += sh2[t + off]; }
    __syncthreads();
  }
  if (t == 0) {
    float m = sh[0] * (1.0f / 16384.0f);
    float var = sh2[0] * (1.0f / 16384.0f) - m * m;
    meanv[bg] = m;
    rstdv[bg] = rsqrtf(var + 1e-5f);
  }
}

// ---------------------------------------------------------------------------
// Apply GroupNorm elementwise: xn = (x - mean)*rstd*gamma + beta
// ---------------------------------------------------------------------------
__global__ void __launch_bounds__(256) gn_apply(const float* __restrict__ x,
                                                const float* __restrict__ meanv,
                                                const float* __restrict__ rstdv,
                                                const float* __restrict__ gamma,
                                                const float* __restrict__ beta,
                                                float* __restrict__ xn) {
  size_t i = (size_t)blockIdx.x * 256 + threadIdx.x;
  int c = (int)(i & (C_ - 1));
  int b = (int)(i >> 19);                 // L_*C_ = 2^19
  int bg = b * 32 + (c >> 4);
  xn[i] = (x[i] - meanv[bg]) * rstdv[bg] * gamma[c] + beta[c];
}

// ---------------------------------------------------------------------------
// GEMM tiles: workgroup 256 threads = 8 waves, tile 256(M) x 64(N).
// Each wave: 32 rows x 64 cols = 8 WMMA C-tiles.  Double-buffered LDS with
// one barrier per k-step; A tile copied by async global->LDS DMA, B tile
// pair-interleaved via a 4-register path stored immediately after the
// barrier (target buffer is dead at that point) so nothing stays live
// across the WMMA block.
// ---------------------------------------------------------------------------

// QKV: A[16384x512] @ W[512x1536] + bias -> q/k/v [B,NH,L,64], factor on q,k
__global__ void __launch_bounds__(256) gemm_qkv(const float* __restrict__ A,
                                                const float* __restrict__ W,
                                                const float* __restrict__ bias,
                                                float* __restrict__ qb_,
                                                float* __restrict__ kb_,
                                                float* __restrict__ vb_) {
  __shared__ float As[2][256 * 20];       // padded stride 20
  __shared__ float Bsp[2][1280];          // pair-interleaved, kp-stride 160
  const int tid = threadIdx.x;
  const int wave = tid >> 5, lane = tid & 31;
  const int lm = lane & 15, lh = lane >> 4;
  const int m0 = blockIdx.x * 256;
  const int n0 = blockIdx.y * 64;
  v8f acc[2][4] = {};

  const int arow = tid >> 2, acc4 = (tid & 3) * 4;   // A tile: 1 chunk x4 rows
  const int brow = tid >> 4, bcol = (tid & 15) * 4;
  const int bbase = (brow >> 1) * 160 + (brow & 1);

  // prologue: tile k0 = 0
#pragma unroll
  for (int i = 0; i < 4; ++i)
    copy16_g2l(A + (size_t)(m0 + arow + 64 * i) * C_ + acc4,
               &As[0][(arow + 64 * i) * 20 + acc4]);
  {
    float4 br = *(const float4*)(W + (size_t)brow * C3_ + n0 + bcol);
#pragma unroll
    for (int u = 0; u < 4; ++u)
      Bsp[0][bbase + (bcol + u) * 2] = ((const float*)&br)[u];
  }
  async_wait_all();

  for (int k0 = 0; k0 < C_; k0 += 16) {
    const int cur = (k0 >> 4) & 1;
    __syncthreads();
    if (k0 + 16 < C_) {                    // stage next tile into other buffer
#pragma unroll
      for (int i = 0; i < 4; ++i)
        copy16_g2l(A + (size_t)(m0 + arow + 64 * i) * C_ + k0 + 16 + acc4,
                   &As[cur ^ 1][(arow + 64 * i) * 20 + acc4]);
      float4 br = *(const float4*)(W + (size_t)(k0 + 16 + brow) * C3_ + n0 + bcol);
#pragma unroll
      for (int u = 0; u < 4; ++u)
        Bsp[cur ^ 1][bbase + (bcol + u) * 2] = ((const float*)&br)[u];
    }
#pragma unroll
    for (int kk = 0; kk < 4; ++kk) {
      v2f a0 = *(const v2f*)(&As[cur][(wave * 32 + lm) * 20 + 4 * kk + 2 * lh]);
      v2f a1 = *(const v2f*)(&As[cur][(wave * 32 + 16 + lm) * 20 + 4 * kk + 2 * lh]);
#pragma unroll
      for (int j = 0; j < 4; ++j) {
        v2f b = *(const v2f*)(&Bsp[cur][(2 * kk + lh) * 160 + (16 * j + lm) * 2]);
        acc[0][j] = wmma4(a0, b, acc[0][j]);
        acc[1][j] = wmma4(a1, b, acc[1][j]);
      }
    }
    async_wait_all();
  }

  const int sel = n0 >> 9;                 // 0=q 1=k 2=v
  const int h = (n0 >> 6) & 7;
  float* dst = (sel == 0) ? qb_ : ((sel == 1) ? kb_ : vb_);
  const float fac = (sel < 2) ? FACTOR : 1.0f;
#pragma unroll
  for (int i = 0; i < 2; ++i)
#pragma unroll
    for (int j = 0; j < 4; ++j) {
      const int d = 16 * j + lm;
      const float bb = bias[n0 + d];
#pragma unroll
      for (int r = 0; r < 8; ++r) {
        int mrow = m0 + wave * 32 + 16 * i + r + 8 * lh;
        int b = mrow >> 10, li = mrow & 1023;
        dst[((size_t)(b * NH_ + h) * L_ + li) * D_ + d] =
            (acc[i][j][r] + bb) * fac;
      }
    }
}

// Out projection: A[16384x512] @ W[512x512] + bias + residual(x) -> out
__global__ void __launch_bounds__(256) gemm_out(const float* __restrict__ A,
                                                const float* __restrict__ W,
                                                const float* __restrict__ bias,
                                                const float* __restrict__ xres,
                                                float* __restrict__ out) {
  __shared__ float As[2][256 * 20];
  __shared__ float Bsp[2][1280];
  const int tid = threadIdx.x;
  const int wave = tid >> 5, lane = tid & 31;
  const int lm = lane & 15, lh = lane >> 4;
  const int m0 = blockIdx.x * 256;
  const int n0 = blockIdx.y * 64;
  v8f acc[2][4] = {};

  const int arow = tid >> 2, acc4 = (tid & 3) * 4;
  const int brow = tid >> 4, bcol = (tid & 15) * 4;
  const int bbase = (brow >> 1) * 160 + (brow & 1);

#pragma unroll
  for (int i = 0; i < 4; ++i)
    copy16_g2l(A + (size_t)(m0 + arow + 64 * i) * C_ + acc4,
               &As[0][(arow + 64 * i) * 20 + acc4]);
  {
    float4 br = *(const float4*)(W + (size_t)brow * C_ + n0 + bcol);
#pragma unroll
    for (int u = 0; u < 4; ++u)
      Bsp[0][bbase + (bcol + u) * 2] = ((const float*)&br)[u];
  }
  async_wait_all();

  for (int k0 = 0; k0 < C_; k0 += 16) {
    const int cur = (k0 >> 4) & 1;
    __syncthreads();
    if (k0 + 16 < C_) {
#pragma unroll
      for (int i = 0; i < 4; ++i)
        copy16_g2l(A + (size_t)(m0 + arow + 64 * i) * C_ + k0 + 16 + acc4,
                   &As[cur ^ 1][(arow + 64 * i) * 20 + acc4]);
      float4 br = *(const float4*)(W + (size_t)(k0 + 16 + brow) * C_ + n0 + bcol);
#pragma unroll
      for (int u = 0; u < 4; ++u)
        Bsp[cur ^ 1][bbase + (bcol + u) * 2] = ((const float*)&br)[u];
    }
#pragma unroll
    for (int kk = 0; kk < 4; ++kk) {
      v2f a0 = *(const v2f*)(&As[cur][(wave * 32 + lm) * 20 + 4 * kk + 2 * lh]);
      v2f a1 = *(const v2f*)(&As[cur][(wave * 32 + 16 + lm) * 20 + 4 * kk + 2 * lh]);
#pragma unroll
      for (int j = 0; j < 4; ++j) {
        v2f b = *(const v2f*)(&Bsp[cur][(2 * kk + lh) * 160 + (16 * j + lm) * 2]);
        acc[0][j] = wmma4(a0, b, acc[0][j]);
        acc[1][j] = wmma4(a1, b, acc[1][j]);
      }
    }
    async_wait_all();
  }

#pragma unroll
  for (int i = 0; i < 2; ++i)
#pragma unroll
    for (int j = 0; j < 4; ++j) {
      const int col = n0 + 16 * j + lm;
      const float bb = bias[col];
#pragma unroll
      for (int r = 0; r < 8; ++r) {
        int mrow = m0 + wave * 32 + 16 * i + r + 8 * lh;
        size_t idx = (size_t)mrow * C_ + col;
        out[idx] = acc[i][j][r] + bb + xres[idx];
      }
    }
}

// ---------------------------------------------------------------------------
// Flash attention, one workgroup per (b*nh, 128 q-rows).  Each wave owns 16
// q-rows; Q tile held in A-fragments throughout.  Double-buffered K/V tiles,
// one barrier per kv-tile; K tile copied via async global->LDS DMA, V tile
// pair-interleaved via a 4-register path stored right after the barrier.
// ---------------------------------------------------------------------------
__global__ void __launch_bounds__(256) attn(const float* __restrict__ Q,
                                            const float* __restrict__ K,
                                            const float* __restrict__ V,
                                            float* __restrict__ O) {
  __shared__ float kt[2][16 * 68];        // row-major, padded stride 68
  __shared__ float vtp[2][1280];          // pair-interleaved, kp-stride 160
  __shared__ float pb[8][16][18];         // per-wave P staging, padded
  const int tid = threadIdx.x;
  const int wave = tid >> 5, lane = tid & 31;
  const int lm = lane & 15, lh = lane >> 4;
  const int bh = blockIdx.y;
  const int q0 = blockIdx.x * 128 + wave * 16;
  const float* Qp = Q + (size_t)bh * L_ * D_;
  const float* Kp = K + (size_t)bh * L_ * D_;
  const float* Vp = V + (size_t)bh * L_ * D_;

  v2f qa[16];
#pragma unroll
  for (int f = 0; f < 16; ++f)
    qa[f] = *(const v2f*)(Qp + (size_t)(q0 + lm) * D_ + 4 * f + 2 * lh);

  v8f o[4] = {};
  float mrun[8], lrun[8];
#pragma unroll
  for (int r = 0; r < 8; ++r) { mrun[r] = -3.0e38f; lrun[r] = 0.0f; }

  const int crow = tid >> 4, ccol = (tid & 15) * 4;
  const int vbase = (crow >> 1) * 160 + (crow & 1);

  // prologue: tile 0
  copy16_g2l(Kp + (size_t)crow * D_ + ccol, &kt[0][crow * 68 + ccol]);
  {
    float4 vr = *(const float4*)(Vp + (size_t)crow * D_ + ccol);
#pragma unroll
    for (int u = 0; u < 4; ++u)
      vtp[0][vbase + (ccol + u) * 2] = ((const float*)&vr)[u];
  }
  async_wait_all();

  for (int kb = 0; kb < 64; ++kb) {
    const int cur = kb & 1;
    __syncthreads();
    if (kb < 63) {                         // stage next K/V tile
      copy16_g2l(Kp + (size_t)((kb + 1) * 16 + crow) * D_ + ccol,
                 &kt[cur ^ 1][crow * 68 + ccol]);
      float4 vr = *(const float4*)(Vp + (size_t)((kb + 1) * 16 + crow) * D_ + ccol);
#pragma unroll
      for (int u = 0; u < 4; ++u)
        vtp[cur ^ 1][vbase + (ccol + u) * 2] = ((const float*)&vr)[u];
    }

    // S(16x16) = Q(16x64) @ K^T, two accumulator chains for ILP
    v8f s0 = {}, s1 = {};
#pragma unroll
    for (int f = 0; f < 16; f += 2) {
      v2f b0 = *(const v2f*)(&kt[cur][lm * 68 + 4 * f + 2 * lh]);
      v2f b1 = *(const v2f*)(&kt[cur][lm * 68 + 4 * (f + 1) + 2 * lh]);
      s0 = wmma4(qa[f], b0, s0);
      s1 = wmma4(qa[f + 1], b1, s1);
    }
    v8f s = s0 + s1;

    // online softmax (rows r / r+8 live in lane halves; reduce over 16 lanes)
    float rm[8];
#pragma unroll
    for (int r = 0; r < 8; ++r) rm[r] = s[r];
#pragma unroll
    for (int msk = 1; msk <= 8; msk <<= 1) {
#pragma unroll
      for (int r = 0; r < 8; ++r)
        rm[r] = fmaxf(rm[r], __shfl_xor(rm[r], msk, 32));
    }
    float p[8], rs[8], alpha[8];
#pragma unroll
    for (int r = 0; r < 8; ++r) {
      float mn = fmaxf(mrun[r], rm[r]);
      p[r] = __expf(s[r] - mn);
      rs[r] = p[r];
      alpha[r] = __expf(mrun[r] - mn);
      mrun[r] = mn;
    }
#pragma unroll
    for (int msk = 1; msk <= 8; msk <<= 1) {
#pragma unroll
      for (int r = 0; r < 8; ++r) rs[r] += __shfl_xor(rs[r], msk, 32);
    }
#pragma unroll
    for (int r = 0; r < 8; ++r) lrun[r] = lrun[r] * alpha[r] + rs[r];
#pragma unroll
    for (int j = 0; j < 4; ++j)
#pragma unroll
      for (int r = 0; r < 8; ++r) o[j][r] *= alpha[r];

    // C-layout -> A-layout for P via wave-private LDS
#pragma unroll
    for (int r = 0; r < 8; ++r) pb[wave][r + 8 * lh][lm] = p[r];
#pragma unroll
    for (int f2 = 0; f2 < 4; ++f2) {
      v2f a = *(const v2f*)(&pb[wave][lm][4 * f2 + 2 * lh]);
#pragma unroll
      for (int j = 0; j < 4; ++j) {
        v2f bv =
            *(const v2f*)(&vtp[cur][(2 * f2 + lh) * 160 + (16 * j + lm) * 2]);
        o[j] = wmma4(a, bv, o[j]);
      }
    }
    async_wait_all();
  }

  const int b = bh >> 3, h = bh & 7;
  float inv[8];
#pragma unroll
  for (int r = 0; r < 8; ++r) inv[r] = 1.0f / lrun[r];
#pragma unroll
  for (int j = 0; j < 4; ++j)
#pragma unroll
    for (int r = 0; r < 8; ++r) {
      int row = q0 + r + 8 * lh;
      O[((size_t)(b * L_ + row)) * C_ + h * D_ + 16 * j + lm] = o[j][r] * inv[r];
    }
}

// ---------------------------------------------------------------------------
extern "C" void kernel_launch(void* const* d_in, const int* in_sizes, int n_in,
                              void* d_out, int out_size, void* d_ws,
                              size_t ws_size, hipStream_t stream) {
  const float* x     = (const float*)d_in[0];
  /* d_in[1] = timesteps (unused by the reference) */
  const float* gamma = (const float*)d_in[2];
  const float* beta  = (const float*)d_in[3];
  const float* w_qkv = (const float*)d_in[4];
  const float* b_qkv = (const float*)d_in[5];
  const float* w_out = (const float*)d_in[6];
  const float* b_out = (const float*)d_in[7];
  float* out = (float*)d_out;

  float* ws = (float*)d_ws;
  float* meanv = ws;                     // 512
  float* rstdv = ws + 512;               // 512
  float* xn = ws + 1024;                 // B*L*C; reused as attention output
  const size_t NEL = (size_t)B_ * L_ * C_;
  float* qb = xn + NEL;
  float* kb = qb + NEL;
  float* vb = kb + NEL;

  gn_stats<<<B_ * G_, 256, 0, stream>>>(x, meanv, rstdv);
  gn_apply<<<(unsigned)(NEL / 256), 256, 0, stream>>>(x, meanv, rstdv, gamma,
                                                      beta, xn);
  gemm_qkv<<<dim3(64, 24), 256, 0, stream>>>(xn, w_qkv, b_qkv, qb, kb, vb);
  attn<<<dim3(8, 128), 256, 0, stream>>>(qb, kb, vb, xn);
  gemm_out<<<dim3(64, 8), 256, 0, stream>>>(xn, w_out, b_out, x, out);
}